// iterativeGCN_mol_40845138985158
// MI455X (gfx1250) — compile-verified
//
#include <hip/hip_runtime.h>

typedef float v2f __attribute__((ext_vector_type(2)));
typedef float v8f __attribute__((ext_vector_type(8)));

#define HID 128
#define BN_EPS 1e-5f

// ---------------------------------------------------------------- encoders
__global__ void atom_encode_kernel(const int* __restrict__ x,
                                   const float* __restrict__ emb,
                                   float* __restrict__ h, int N) {
  long long g = (long long)blockIdx.x * blockDim.x + threadIdx.x;
  int n = (int)(g >> 5), lane = (int)(g & 31);
  if (n >= N) return;
  const int off[9] = {0, 119, 123, 135, 147, 157, 163, 169, 171};
  float4 acc = make_float4(0.f, 0.f, 0.f, 0.f);
#pragma unroll
  for (int f = 0; f < 9; ++f) {
    int idx = x[n * 9 + f] + off[f];
    float4 v = ((const float4*)(emb + (long long)idx * HID))[lane];
    acc.x += v.x; acc.y += v.y; acc.z += v.z; acc.w += v.w;
  }
  ((float4*)(h + (long long)n * HID))[lane] = acc;
}

__global__ void bond_encode_kernel(const int* __restrict__ ea,
                                   const float* __restrict__ emb,
                                   float* __restrict__ e, int E) {
  long long g = (long long)blockIdx.x * blockDim.x + threadIdx.x;
  int n = (int)(g >> 5), lane = (int)(g & 31);
  if (n >= E) return;
  const int off[3] = {0, 5, 11};
  float4 acc = make_float4(0.f, 0.f, 0.f, 0.f);
#pragma unroll
  for (int f = 0; f < 3; ++f) {
    int idx = ea[n * 3 + f] + off[f];
    float4 v = ((const float4*)(emb + (long long)idx * HID))[lane];
    acc.x += v.x; acc.y += v.y; acc.z += v.z; acc.w += v.w;
  }
  ((float4*)(e + (long long)n * HID))[lane] = acc;
}

// ---------------------------------------------------------------- degree / norm
__global__ void deg_count_kernel(const int* __restrict__ row, float* __restrict__ deg, int E) {
  int i = blockIdx.x * blockDim.x + threadIdx.x;
  if (i < E) atomicAdd(&deg[row[i]], 1.0f);
}

__global__ void deg_finalize_kernel(float* __restrict__ deg, float* __restrict__ dis,
                                    float* __restrict__ invd, int N) {
  int i = blockIdx.x * blockDim.x + threadIdx.x;
  if (i >= N) return;
  float d = deg[i] + 1.0f;
  dis[i]  = 1.0f / sqrtf(d);
  invd[i] = 1.0f / d;
}

__global__ void edge_norm_kernel(const int* __restrict__ row, const int* __restrict__ col,
                                 const float* __restrict__ dis, float* __restrict__ norm, int E) {
  int i = blockIdx.x * blockDim.x + threadIdx.x;
  if (i < E) norm[i] = dis[row[i]] * dis[col[i]];
}

// ---------------------------------------------------------------- WMMA GEMM
// out[M,128] = A[M,128] @ W^T + bias, W is [128 out][128 in] row-major (f32).
// Per wave: 16 rows x 128 cols, K=128 as 32 steps of v_wmma_f32_16x16x4_f32
// per N-tile. Fragment loads for step k+1 are issued before the WMMA chain of
// step k (software pipelining) so ds waits overlap with matrix work.
// LDS pitch 132 floats -> conflict-free b64 fragment reads.
__device__ __forceinline__ void load_frags(const float* __restrict__ Arow,
                                           const float* __restrict__ Wl,
                                           int lr, int koff, int k,
                                           v2f& a, v2f (&b)[8]) {
  a[0] = Arow[4 * k + koff];
  a[1] = Arow[4 * k + koff + 1];
#pragma unroll
  for (int nt = 0; nt < 8; ++nt) {
    b[nt][0] = Wl[(nt * 16 + lr) * 132 + 4 * k + koff];      // B[k][n] = W[n][k]
    b[nt][1] = Wl[(nt * 16 + lr) * 132 + 4 * k + koff + 1];
  }
}

__launch_bounds__(128)
__global__ void gemm128_wmma(const float* __restrict__ A, const float* __restrict__ W,
                             const float* __restrict__ bias, float* __restrict__ out, int M) {
  extern __shared__ float lds[];
  float* Wl = lds;               // 128 x 132
  float* Al = lds + 128 * 132;   // 64 x 132
  const int t    = threadIdx.x;
  const int lane = t & 31;
  const int wave = t >> 5;
  const int lr   = lane & 15;
  const int koff = (lane < 16) ? 0 : 2;

  // stage W once per block (float4 coalesced)
  for (int i = t; i < 128 * 32; i += 128) {
    int r = i >> 5, c4 = (i & 31) << 2;
    float4 v = ((const float4*)W)[i];
    Wl[r * 132 + c4 + 0] = v.x; Wl[r * 132 + c4 + 1] = v.y;
    Wl[r * 132 + c4 + 2] = v.z; Wl[r * 132 + c4 + 3] = v.w;
  }
  float bv[8];
#pragma unroll
  for (int nt = 0; nt < 8; ++nt) bv[nt] = bias[nt * 16 + lr];

  const int nChunks = (M + 63) >> 6;
  for (int chunk = blockIdx.x; chunk < nChunks; chunk += gridDim.x) {
    __syncthreads();  // covers W-stage (1st iter) & previous chunk's reads of Al
    const int base = chunk << 6;
    for (int i = t; i < 64 * 32; i += 128) {
      int r = i >> 5, c4 = (i & 31) << 2;
      int row = base + r;
      float4 v = make_float4(0.f, 0.f, 0.f, 0.f);
      if (row < M) v = ((const float4*)(A + (long long)row * HID))[i & 31];
      Al[r * 132 + c4 + 0] = v.x; Al[r * 132 + c4 + 1] = v.y;
      Al[r * 132 + c4 + 2] = v.z; Al[r * 132 + c4 + 3] = v.w;
    }
    __syncthreads();

    const int row0 = base + wave * 16;
    if (row0 < M) {  // wave-uniform guard: EXEC stays all-1s around WMMA
      v8f acc[8];
#pragma unroll
      for (int nt = 0; nt < 8; ++nt) {
#pragma unroll
        for (int j = 0; j < 8; ++j) acc[nt][j] = bv[nt];
      }
      const float* Arow = Al + (wave * 16 + lr) * 132;

      v2f a, b[8];
      load_frags(Arow, Wl, lr, koff, 0, a, b);
#pragma unroll 4
      for (int k = 0; k < 31; ++k) {
        v2f an, bn[8];
        load_frags(Arow, Wl, lr, koff, k + 1, an, bn);   // prefetch next step
#pragma unroll
        for (int nt = 0; nt < 8; ++nt)
          acc[nt] = __builtin_amdgcn_wmma_f32_16x16x4_f32(
              false, a, false, b[nt], (short)0, acc[nt], false, false);
        a = an;
#pragma unroll
        for (int nt = 0; nt < 8; ++nt) b[nt] = bn[nt];
      }
#pragma unroll
      for (int nt = 0; nt < 8; ++nt)                      // epilogue step k=31
        acc[nt] = __builtin_amdgcn_wmma_f32_16x16x4_f32(
            false, a, false, b[nt], (short)0, acc[nt], false, false);

      const int mo = (lane < 16) ? 0 : 8;
#pragma unroll
      for (int nt = 0; nt < 8; ++nt) {
#pragma unroll
        for (int j = 0; j < 8; ++j)
          out[(long long)(row0 + j + mo) * HID + nt * 16 + lr] = acc[nt][j];
      }
    }
  }
}

// ---------------------------------------------------------------- edge message + scatter
__global__ void edge_msg_kernel(const float* __restrict__ z, const float* __restrict__ e,
                                const float* __restrict__ norm, const int* __restrict__ row,
                                const int* __restrict__ col, float* __restrict__ agg, int E) {
  long long g = (long long)blockIdx.x * blockDim.x + threadIdx.x;
  int edge = (int)(g >> 5), lane = (int)(g & 31);
  if (edge >= E) return;
  int r = row[edge], c = col[edge];
  float nv = norm[edge];
  float4 zv = ((const float4*)(z + (long long)r * HID))[lane];
  float4 ev = ((const float4*)(e + (long long)edge * HID))[lane];
  float* ap = agg + (long long)c * HID + lane * 4;
  atomicAdd(ap + 0, fmaxf(zv.x + ev.x, 0.f) * nv);
  atomicAdd(ap + 1, fmaxf(zv.y + ev.y, 0.f) * nv);
  atomicAdd(ap + 2, fmaxf(zv.z + ev.z, 0.f) * nv);
  atomicAdd(ap + 3, fmaxf(zv.w + ev.w, 0.f) * nv);
}

// ---------------------------------------------------------------- self-loop + relu + BN + blend
__global__ void combine_kernel(float* __restrict__ h, const float* __restrict__ z,
                               const float* __restrict__ agg, const float* __restrict__ root,
                               const float* __restrict__ invd, const float* __restrict__ gamma,
                               const float* __restrict__ beta, const float* __restrict__ mean,
                               const float* __restrict__ var, float fi, long long total) {
  long long i = (long long)blockIdx.x * blockDim.x + threadIdx.x;
  if (i >= total) return;
  int n = (int)(i >> 7), c = (int)(i & 127);
  float v = agg[i] + fmaxf(z[i] + root[c], 0.f) * invd[n];
  v = fmaxf(v, 0.f);
  float inv_std = gamma[c] / sqrtf(var[c] + BN_EPS);
  v = (v - mean[c]) * inv_std + beta[c];
  h[i] = fi * h[i] + (1.0f - fi) * v;
}

// ---------------------------------------------------------------- pooling
__global__ void pool_acc_kernel(const float* __restrict__ h, const int* __restrict__ batch,
                                float* __restrict__ pooled, float* __restrict__ counts, int N) {
  long long g = (long long)blockIdx.x * blockDim.x + threadIdx.x;
  int n = (int)(g >> 5), lane = (int)(g & 31);
  if (n >= N) return;
  int grp = batch[n];
  if (lane == 0) atomicAdd(&counts[grp], 1.0f);
  float4 hv = ((const float4*)(h + (long long)n * HID))[lane];
  float* pp = pooled + (long long)grp * HID + lane * 4;
  atomicAdd(pp + 0, hv.x); atomicAdd(pp + 1, hv.y);
  atomicAdd(pp + 2, hv.z); atomicAdd(pp + 3, hv.w);
}

__global__ void pool_norm_kernel(float* __restrict__ pooled, const float* __restrict__ counts,
                                 long long total) {
  long long i = (long long)blockIdx.x * blockDim.x + threadIdx.x;
  if (i >= total) return;
  pooled[i] /= fmaxf(counts[i >> 7], 1.0f);
}

// ---------------------------------------------------------------- host
extern "C" void kernel_launch(void* const* d_in, const int* in_sizes, int n_in,
                              void* d_out, int out_size, void* d_ws, size_t ws_size,
                              hipStream_t stream) {
  const int*   x         = (const int*)d_in[0];
  const int*   edge_index= (const int*)d_in[1];
  const int*   edge_attr = (const int*)d_in[2];
  const int*   batch     = (const int*)d_in[3];
  const float* atom_emb  = (const float*)d_in[4];
  const float* bond_emb  = (const float*)d_in[5];
  const float* conv_W    = (const float*)d_in[6];
  const float* conv_b    = (const float*)d_in[7];
  const float* root_emb  = (const float*)d_in[8];
  const float* bn_gamma  = (const float*)d_in[9];
  const float* bn_beta   = (const float*)d_in[10];
  const float* bn_mean   = (const float*)d_in[11];
  const float* bn_var    = (const float*)d_in[12];
  const float* lin_W     = (const float*)d_in[13];
  const float* lin_b     = (const float*)d_in[14];

  const int N = in_sizes[0] / 9;      // 300000
  const int E = in_sizes[2] / 3;      // 600000
  const int G = 12000;
  const int* row = edge_index;
  const int* col = edge_index + E;

  // carve workspace
  char* ws = (char*)d_ws;
  size_t off = 0;
  auto carve = [&](size_t bytes) -> void* {
    void* p = ws + off;
    off += (bytes + 255) & ~(size_t)255;
    return p;
  };
  float* h      = (float*)carve((size_t)N * HID * 4);
  float* z      = (float*)carve((size_t)N * HID * 4);
  float* agg    = (float*)carve((size_t)N * HID * 4);
  float* e      = (float*)carve((size_t)E * HID * 4);
  float* deg    = (float*)carve((size_t)N * 4);
  float* dis    = (float*)carve((size_t)N * 4);
  float* invd   = (float*)carve((size_t)N * 4);
  float* norm   = (float*)carve((size_t)E * 4);
  float* pooled = (float*)carve((size_t)G * HID * 4);
  float* counts = (float*)carve((size_t)G * 4);

  const int B = 256;
  const size_t gemm_lds = (size_t)(128 * 132 + 64 * 132) * sizeof(float);  // 101376 B

  // encoders
  atom_encode_kernel<<<((long long)N * 32 + B - 1) / B, B, 0, stream>>>(x, atom_emb, h, N);
  bond_encode_kernel<<<((long long)E * 32 + B - 1) / B, B, 0, stream>>>(edge_attr, bond_emb, e, E);

  // degree / norm
  hipMemsetAsync(deg, 0, (size_t)N * 4, stream);
  deg_count_kernel<<<(E + B - 1) / B, B, 0, stream>>>(row, deg, E);
  deg_finalize_kernel<<<(N + B - 1) / B, B, 0, stream>>>(deg, dis, invd, N);
  edge_norm_kernel<<<(E + B - 1) / B, B, 0, stream>>>(row, col, dis, norm, E);

  // message-passing rounds. i==1 computes h = 1*old + 0*z == old exactly -> skip.
  const int nChunksN = (N + 63) / 64;
  const int gemmGridN = nChunksN < 1024 ? nChunksN : 1024;
  for (int i = 0; i < 5; ++i) {
    if (i == 1) continue;
    gemm128_wmma<<<gemmGridN, 128, gemm_lds, stream>>>(h, conv_W, conv_b, z, N);
    hipMemsetAsync(agg, 0, (size_t)N * HID * 4, stream);
    edge_msg_kernel<<<((long long)E * 32 + B - 1) / B, B, 0, stream>>>(z, e, norm, row, col, agg, E);
    combine_kernel<<<((long long)N * HID + B - 1) / B, B, 0, stream>>>(
        h, z, agg, root_emb, invd, bn_gamma, bn_beta, bn_mean, bn_var,
        (float)i, (long long)N * HID);
  }

  // global mean pool + final linear (WMMA) into d_out
  hipMemsetAsync(pooled, 0, (size_t)G * HID * 4, stream);
  hipMemsetAsync(counts, 0, (size_t)G * 4, stream);
  pool_acc_kernel<<<((long long)N * 32 + B - 1) / B, B, 0, stream>>>(h, batch, pooled, counts, N);
  pool_norm_kernel<<<((long long)G * HID + B - 1) / B, B, 0, stream>>>(pooled, counts, (long long)G * HID);

  const int nChunksG = (G + 63) / 64;
  const int gemmGridG = nChunksG < 1024 ? nChunksG : 1024;
  gemm128_wmma<<<gemmGridG, 128, gemm_lds, stream>>>(pooled, lin_W, lin_b, (float*)d_out, G);
}